// SimplifiedDiffusionModel_11982958756146
// MI455X (gfx1250) — compile-verified
//
#include <hip/hip_runtime.h>

typedef float v2f __attribute__((ext_vector_type(2)));
typedef float v8f __attribute__((ext_vector_type(8)));

#define DIMD 64
#define NZ   4096
#define NT   8192
#define NE   65536
#define H1   128
#define H2   64
#define ZW   (NZ / 32)   // 128 mask words per adjacency row

// ---------------------------------------------------------------- adjacency
__global__ __launch_bounds__(256) void k_zero_adj(uint4* adj4) {
    int i = blockIdx.x * 256 + threadIdx.x;   // 131072 uint4 = 2MB
    adj4[i] = make_uint4(0u, 0u, 0u, 0u);
}

__global__ __launch_bounds__(256) void k_scatter_adj(const int* __restrict__ ei,
                                                     unsigned* __restrict__ adj) {
    int i = blockIdx.x * 256 + threadIdx.x;
    if (i < NE) {
        int u = ei[i];
        int v = ei[NE + i];
        atomicOr(&adj[(size_t)u * ZW + (v >> 5)], 1u << (v & 31));
        atomicOr(&adj[(size_t)v * ZW + (u >> 5)], 1u << (u & 31));
    } else if (i < NE + NZ) {
        int z = i - NE;
        atomicOr(&adj[(size_t)z * ZW + (z >> 5)], 1u << (z & 31));
    }
}

// ---------------------------------------------------------------- tiny MLP: h2[T,64]
__global__ __launch_bounds__(256) void k_mlp(const float* __restrict__ pa,
                                             const float* __restrict__ times,
                                             const float* __restrict__ W1,
                                             const float* __restrict__ b1,
                                             const float* __restrict__ W2,
                                             const float* __restrict__ b2,
                                             float* __restrict__ h2out) {
    __shared__ float h1s[32 * H1];   // 16 KB
    __shared__ float pas[DIMD];
    const int tid = threadIdx.x;
    const int t0  = blockIdx.x * 32;
    if (tid < DIMD) pas[tid] = pa[tid];
    __syncthreads();

    // layer 1: x = [person_attrs, time]  ->  relu(x @ W1 + b1)
    for (int o = tid; o < 32 * H1; o += 256) {
        int tt = o >> 7, j = o & (H1 - 1);
        float acc = b1[j];
        #pragma unroll 8
        for (int d = 0; d < DIMD; ++d) acc = fmaf(pas[d], W1[d * H1 + j], acc);
        acc = fmaf(times[t0 + tt], W1[DIMD * H1 + j], acc);
        h1s[tt * H1 + j] = fmaxf(acc, 0.0f);
    }
    __syncthreads();

    // layer 2: relu(h1 @ W2 + b2)
    for (int o = tid; o < 32 * H2; o += 256) {
        int tt = o >> 6, j = o & (H2 - 1);
        float acc = b2[j];
        #pragma unroll 8
        for (int k = 0; k < H1; ++k) acc = fmaf(h1s[tt * H1 + k], W2[k * H2 + j], acc);
        h2out[(size_t)(t0 + tt) * H2 + j] = fmaxf(acc, 0.0f);
    }
}

// ---------------------------------------------------------------- WMMA GEMM: raw = h2 @ W3 + b3
// block tile 16(M) x 128(N), K=64; 8 waves, one 16x16 WMMA tile per wave, 16 k-steps of 4.
__global__ __launch_bounds__(256) void k_gemm(const float* __restrict__ A,    // [T,64]
                                              const float* __restrict__ B,    // [64,Z]
                                              const float* __restrict__ bias, // [Z]
                                              float* __restrict__ C) {        // [T,Z]
    __shared__ float As[16 * 64];    //  4 KB
    __shared__ float Bs[64 * 128];   // 32 KB
    const int tid  = threadIdx.x;
    const int m0   = blockIdx.x * 16;
    const int n0   = blockIdx.y * 128;

    // A tile: 16 rows x 64 cols, contiguous (row-major [T,64]) -> 256 float4, one per thread
    ((float4*)As)[tid] = ((const float4*)(A + (size_t)m0 * 64))[tid];
    // B tile: 64 rows x 128 cols -> 2048 float4, 8 per thread
    for (int i = tid; i < 64 * 32; i += 256) {
        int k = i >> 5, c4 = i & 31;
        ((float4*)Bs)[i] = *(const float4*)(B + (size_t)k * NZ + n0 + c4 * 4);
    }
    __syncthreads();

    const int wave = tid >> 5;
    const int lane = tid & 31;
    const int half = lane >> 4;       // 0: k,k+1 | 1: k+2,k+3
    const int lm   = lane & 15;
    const int nloc = wave * 16 + lm;  // column within block tile

    v8f acc = {};
    #pragma unroll
    for (int kk = 0; kk < 16; ++kk) {
        const int k = kk * 4 + 2 * half;
        v2f a, b;
        a[0] = As[lm * 64 + k];
        a[1] = As[lm * 64 + k + 1];
        b[0] = Bs[k * 128 + nloc];
        b[1] = Bs[(k + 1) * 128 + nloc];
        acc = __builtin_amdgcn_wmma_f32_16x16x4_f32(false, a, false, b,
                                                    (short)0, acc, false, false);
    }

    const int col = n0 + nloc;
    const float bv = bias[col];
    #pragma unroll
    for (int r = 0; r < 8; ++r) {
        int row = m0 + half * 8 + r;     // C layout: VGPR r = M r (lanes 0-15), M 8+r (lanes 16-31)
        C[(size_t)row * NZ + col] = acc[r] + bv;
    }
}

// ---------------------------------------------------------------- sequential zone chain
// One workgroup (8 waves). Double-buffered logits rows in LDS: the global loads
// for row t+1 are issued before the reduce of row t, so HBM/L2 latency hides
// behind the argmax. Wave32 shfl reduce: only 3 barriers per step.
__global__ __launch_bounds__(256) void k_chain(const float* __restrict__ lg,   // raw logits [T,Z]
                                               const unsigned* __restrict__ adj,
                                               int* __restrict__ zones) {
    __shared__ float    row[2][NZ];   // 32 KB double buffer
    __shared__ unsigned msk[ZW];      // 512 B
    __shared__ float    wbest[8];
    __shared__ int      wibest[8];
    __shared__ int      curz;
    const int tid  = threadIdx.x;
    const int lane = tid & 31;
    const int wave = tid >> 5;
    if (tid == 0) curz = 0;

    // prologue: stage row 0
    {
        const float4* src = (const float4*)lg;
        #pragma unroll
        for (int j = 0; j < 4; ++j) ((float4*)row[0])[tid + j * 256] = src[tid + j * 256];
    }

    for (int t = 0; t < NT; ++t) {
        const int cur = t & 1;
        const int nxt = cur ^ 1;

        // issue next-row loads into registers NOW (wait lands after the reduce)
        float4 pre[4];
        if (t + 1 < NT) {
            const float4* src = (const float4*)(lg + (size_t)(t + 1) * NZ);
            #pragma unroll
            for (int j = 0; j < 4; ++j) pre[j] = src[tid + j * 256];
        }
        if (t + 2 < NT)
            __builtin_prefetch(lg + (size_t)(t + 2) * NZ + tid * 16, 0, 1);

        // adjacency mask row of current zone (chain-dependent, cannot prefetch)
        if (tid < ZW) msk[tid] = adj[(size_t)curz * ZW + tid];
        __syncthreads();                                   // A: row[cur], msk, curz visible

        // masked argmax: each thread covers 16 zones; first-max-wins (lower index on tie)
        unsigned bits = (msk[tid >> 1] >> ((tid & 1) * 16)) & 0xFFFFu;
        float best = -INFINITY;
        int   bi   = NZ;
        const int base = tid * 16;
        #pragma unroll
        for (int b = 0; b < 16; ++b) {
            if (bits & (1u << b)) {
                float v = row[cur][base + b];
                if (v > best) { best = v; bi = base + b; }
            }
        }
        // wave32 shuffle reduce (no barriers)
        #pragma unroll
        for (int s = 16; s > 0; s >>= 1) {
            float ov = __shfl_down(best, s, 32);
            int   oi = __shfl_down(bi,   s, 32);
            if (ov > best || (ov == best && oi < bi)) { best = ov; bi = oi; }
        }
        if (lane == 0) { wbest[wave] = best; wibest[wave] = bi; }

        // commit prefetched row t+1 to the alternate buffer (loadcnt wait lands here)
        if (t + 1 < NT) {
            #pragma unroll
            for (int j = 0; j < 4; ++j) ((float4*)row[nxt])[tid + j * 256] = pre[j];
        }
        __syncthreads();                                   // B: wbest/wibest visible

        if (tid == 0) {
            float fb = wbest[0];
            int   fi = wibest[0];
            #pragma unroll
            for (int w = 1; w < 8; ++w) {
                if (wbest[w] > fb || (wbest[w] == fb && wibest[w] < fi)) {
                    fb = wbest[w]; fi = wibest[w];
                }
            }
            zones[t] = curz;        // zone used for THIS step's penalty row
            curz = fi;              // carry for next step
        }
        __syncthreads();                                   // C: curz update visible
    }
}

// ---------------------------------------------------------------- parallel penalty apply
__global__ __launch_bounds__(256) void k_penalty(float* __restrict__ outp,
                                                 const unsigned* __restrict__ adj,
                                                 const int* __restrict__ zones) {
    const int t   = blockIdx.x;
    const int tid = threadIdx.x;
    const int zone = zones[t];
    unsigned bits = (adj[(size_t)zone * ZW + (tid >> 1)] >> ((tid & 1) * 16)) & 0xFFFFu;
    float4* o = (float4*)(outp + (size_t)t * NZ);
    #pragma unroll
    for (int j = 0; j < 4; ++j) {
        float4 v = o[tid * 4 + j];
        if (!((bits >> (j * 4 + 0)) & 1u)) v.x -= 10.0f;
        if (!((bits >> (j * 4 + 1)) & 1u)) v.y -= 10.0f;
        if (!((bits >> (j * 4 + 2)) & 1u)) v.z -= 10.0f;
        if (!((bits >> (j * 4 + 3)) & 1u)) v.w -= 10.0f;
        o[tid * 4 + j] = v;
    }
}

// ---------------------------------------------------------------- launch
extern "C" void kernel_launch(void* const* d_in, const int* in_sizes, int n_in,
                              void* d_out, int out_size, void* d_ws, size_t ws_size,
                              hipStream_t stream) {
    const float* pa    = (const float*)d_in[0];   // [64]
    const float* times = (const float*)d_in[1];   // [8192]
    // d_in[2] = zone_features: unused by the reference
    const int*   ei    = (const int*)  d_in[3];   // [2,E]
    const float* W1    = (const float*)d_in[4];   // [65,128]
    const float* b1    = (const float*)d_in[5];   // [128]
    const float* W2    = (const float*)d_in[6];   // [128,64]
    const float* b2    = (const float*)d_in[7];   // [64]
    const float* W3    = (const float*)d_in[8];   // [64,4096]
    const float* b3    = (const float*)d_in[9];   // [4096]
    float* out = (float*)d_out;                   // [T,Z]

    const size_t adj_bytes = (size_t)NZ * ZW * 4;     // 2 MB
    const size_t h2_bytes  = (size_t)NT * H2 * 4;     // 2 MB
    unsigned* adj   = (unsigned*)d_ws;
    float*    h2    = (float*)((char*)d_ws + adj_bytes);
    int*      zones = (int*)((char*)d_ws + adj_bytes + h2_bytes);

    k_zero_adj   <<<(NZ * ZW / 4) / 256, 256, 0, stream>>>((uint4*)adj);
    k_scatter_adj<<<(NE + NZ + 255) / 256, 256, 0, stream>>>(ei, adj);
    k_mlp        <<<NT / 32, 256, 0, stream>>>(pa, times, W1, b1, W2, b2, h2);
    dim3 g(NT / 16, NZ / 128);
    k_gemm       <<<g, 256, 0, stream>>>(h2, W3, b3, out);
    k_chain      <<<1, 256, 0, stream>>>(out, adj, zones);
    k_penalty    <<<NT, 256, 0, stream>>>(out, adj, zones);
}